// EntityEmbedding_28630251995241
// MI455X (gfx1250) — compile-verified
//
#include <hip/hip_runtime.h>

typedef float v2f __attribute__((ext_vector_type(2)));
typedef float v8f __attribute__((ext_vector_type(8)));

#define B_DIM 32
#define S_DIM 4096
#define H_DIM 1024

// ---------------------------------------------------------------------------
// Pass 1: per batch row, find the first s with seg==1 and the adjusted count
// (count of ones minus one, matching the reference's positions[1:] drop).
// ---------------------------------------------------------------------------
__global__ __launch_bounds__(256) void mask_stats_kernel(
    const int* __restrict__ seg,
    int* __restrict__ firstIdx,
    float* __restrict__ invCnt) {
  __shared__ int sFirst[256];
  __shared__ int sCnt[256];
  const int b = blockIdx.x;
  const int tid = threadIdx.x;
  const int* row = seg + (size_t)b * S_DIM;

  int first = S_DIM;
  int cnt = 0;
  for (int s = tid; s < S_DIM; s += 256) {
    if (row[s] == 1) {
      cnt++;
      if (s < first) first = s;
    }
  }
  sFirst[tid] = first;
  sCnt[tid] = cnt;
  __syncthreads();
  for (int off = 128; off > 0; off >>= 1) {
    if (tid < off) {
      sFirst[tid] = min(sFirst[tid], sFirst[tid + off]);
      sCnt[tid] += sCnt[tid + off];
    }
    __syncthreads();
  }
  if (tid == 0) {
    const int f = sFirst[0];
    const int tot = sCnt[0];
    const float c = (f < S_DIM) ? (float)(tot - 1) : (float)tot;
    firstIdx[b] = f;
    invCnt[b] = 1.0f / c;  // c==0 -> inf; 0*inf -> NaN, same as reference 0/0
  }
}

// ---------------------------------------------------------------------------
// Pass 2: masked sum over S via V_WMMA_F32_16X16X4_F32.
//   A[16,4]  = mask values for 4 consecutive s (broadcast over M)
//   B[4,16]  = X tile: 4 s-rows x 16 h-cols (coalesced b32 loads per lane)
//   C        = fp32 accumulator; result for h0+n lives in c[0] of lanes 0..15
// One block = 8 waves, each wave owns a 512-element s-slice of one (b, h-tile).
// ---------------------------------------------------------------------------
__global__ __launch_bounds__(256) void pool_wmma_kernel(
    const float* __restrict__ X,
    const int* __restrict__ seg,
    const int* __restrict__ firstIdx,
    const float* __restrict__ invCnt,
    float* __restrict__ out) {
  __shared__ float part[8][16];

  const int b = blockIdx.x;
  const int h0 = blockIdx.y * 16;
  const int tid = threadIdx.x;
  const int wave = tid >> 5;
  const int lane = tid & 31;
  const int n = lane & 15;          // h column within the tile
  const int kHi = (lane >> 4) << 1; // 0 for lanes 0-15, 2 for lanes 16-31

  const int first = firstIdx[b];
  const int* segRow = seg + (size_t)b * S_DIM;
  const float* Xp = X + (size_t)b * S_DIM * H_DIM + h0 + n;

  v8f c0 = {};
  v8f c1 = {};

  const int sBegin = wave * (S_DIM / 8);
  const int sEnd = sBegin + (S_DIM / 8);

  for (int s = sBegin; s < sEnd; s += 8) {
    // Scalar (wave-uniform) mask values for 8 sequence positions.
    const float m0 = (segRow[s + 0] == 1 && (s + 0) != first) ? 1.0f : 0.0f;
    const float m1 = (segRow[s + 1] == 1 && (s + 1) != first) ? 1.0f : 0.0f;
    const float m2 = (segRow[s + 2] == 1 && (s + 2) != first) ? 1.0f : 0.0f;
    const float m3 = (segRow[s + 3] == 1 && (s + 3) != first) ? 1.0f : 0.0f;
    const float m4 = (segRow[s + 4] == 1 && (s + 4) != first) ? 1.0f : 0.0f;
    const float m5 = (segRow[s + 5] == 1 && (s + 5) != first) ? 1.0f : 0.0f;
    const float m6 = (segRow[s + 6] == 1 && (s + 6) != first) ? 1.0f : 0.0f;
    const float m7 = (segRow[s + 7] == 1 && (s + 7) != first) ? 1.0f : 0.0f;

    // --- chunk 0: s .. s+3 ---
    v2f a0;
    a0.x = kHi ? m2 : m0;
    a0.y = kHi ? m3 : m1;
    const float* p0 = Xp + (size_t)(s + kHi) * H_DIM;
    v2f b0;
    b0.x = p0[0];
    b0.y = p0[H_DIM];
    c0 = __builtin_amdgcn_wmma_f32_16x16x4_f32(false, a0, false, b0,
                                               (short)0, c0, false, false);

    // --- chunk 1: s+4 .. s+7 (independent accumulator) ---
    v2f a1;
    a1.x = kHi ? m6 : m4;
    a1.y = kHi ? m7 : m5;
    const float* p1 = Xp + (size_t)(s + 4 + kHi) * H_DIM;
    v2f b1;
    b1.x = p1[0];
    b1.y = p1[H_DIM];
    c1 = __builtin_amdgcn_wmma_f32_16x16x4_f32(false, a1, false, b1,
                                               (short)0, c1, false, false);
  }

  const float partial = c0[0] + c1[0];
  if (lane < 16) part[wave][n] = partial;
  __syncthreads();

  if (tid < 16) {
    float acc = 0.0f;
#pragma unroll
    for (int w = 0; w < 8; ++w) acc += part[w][tid];
    out[(size_t)b * H_DIM + h0 + tid] = acc * invCnt[b];
  }
}

// ---------------------------------------------------------------------------
extern "C" void kernel_launch(void* const* d_in, const int* in_sizes, int n_in,
                              void* d_out, int out_size, void* d_ws,
                              size_t ws_size, hipStream_t stream) {
  const float* X = (const float*)d_in[0];   // [B, S, H] fp32
  const int* seg = (const int*)d_in[1];     // [B, S] integer

  int* firstIdx = (int*)d_ws;                        // 32 ints
  float* invCnt = (float*)((char*)d_ws + 128);       // 32 floats
  float* out = (float*)d_out;                        // [B, 1, H] fp32

  mask_stats_kernel<<<dim3(B_DIM), dim3(256), 0, stream>>>(seg, firstIdx,
                                                           invCnt);
  pool_wmma_kernel<<<dim3(B_DIM, H_DIM / 16), dim3(256), 0, stream>>>(
      X, seg, firstIdx, invCnt, out);
}